// SparseSoftmax_65944927863275
// MI455X (gfx1250) — compile-verified
//
#include <hip/hip_runtime.h>

// Masked (sparse) softmax over rows of length 2048.
// features: fp32 (32,2048,2048), OD: int32 mask, out: fp32.
// One 256-thread block (8 wave32 waves) per row; each thread owns 8 elements,
// held in registers for the whole kernel (single read pass, single write pass).
// Reductions: ds_swizzle_b32 XOR butterflies (wave32) + 8-slot LDS cross-wave.

#define ROW_LEN   2048
#define TPB       256
#define WAVES     (TPB / 32)          // 8 waves per block (wave32)
#define NEG_INF   (-__builtin_inff())

typedef float  v4f __attribute__((ext_vector_type(4)));
typedef int    v4i __attribute__((ext_vector_type(4)));

// ds_swizzle_b32 group-of-32 XOR exchange: offset = xor<<10 | or<<5 | and
// and_mask=0x1f keeps full 32-lane sharing (wave32 on gfx1250).
template <int XOR>
__device__ __forceinline__ float swz_xor(float v) {
    constexpr int pat = (XOR << 10) | 0x1f;
    return __int_as_float(__builtin_amdgcn_ds_swizzle(__float_as_int(v), pat));
}

__device__ __forceinline__ float wave_reduce_max(float v) {
    v = fmaxf(v, swz_xor<16>(v));
    v = fmaxf(v, swz_xor<8>(v));
    v = fmaxf(v, swz_xor<4>(v));
    v = fmaxf(v, swz_xor<2>(v));
    v = fmaxf(v, swz_xor<1>(v));
    return v;
}

__device__ __forceinline__ float wave_reduce_sum(float v) {
    v += swz_xor<16>(v);
    v += swz_xor<8>(v);
    v += swz_xor<4>(v);
    v += swz_xor<2>(v);
    v += swz_xor<1>(v);
    return v;
}

__global__ __launch_bounds__(TPB) void SparseSoftmax_65944927863275_kernel(
    const float* __restrict__ feat,
    const int*   __restrict__ od,
    float*       __restrict__ out)
{
    __shared__ float red_max[WAVES];
    __shared__ float red_sum[WAVES];

    const int tid  = threadIdx.x;
    const int lane = tid & 31;
    const int wave = tid >> 5;

    const size_t rowBase = (size_t)blockIdx.x * (size_t)ROW_LEN;
    const float* f = feat + rowBase;
    const int*   m = od   + rowBase;
    float*       o = out  + rowBase;

    // Two contiguous 512-byte-per-wave chunks: lanes t*4 .. t*4+3 and +1024.
    const int i0 = tid * 4;
    const int i1 = i0 + (ROW_LEN / 2);

    // ---- Single streaming read of the row (non-temporal: no reuse, >L2) ----
    v4f f0 = __builtin_nontemporal_load((const v4f*)(f + i0));
    v4f f1 = __builtin_nontemporal_load((const v4f*)(f + i1));
    v4i m0 = __builtin_nontemporal_load((const v4i*)(m + i0));
    v4i m1 = __builtin_nontemporal_load((const v4i*)(m + i1));

    // Masked scores (kept in registers)
    float s[8];
    #pragma unroll
    for (int j = 0; j < 4; ++j) {
        s[j]     = (m0[j] != 0) ? f0[j] : NEG_INF;
        s[4 + j] = (m1[j] != 0) ? f1[j] : NEG_INF;
    }

    // ---- Row max (masked) ----
    float lmax = s[0];
    #pragma unroll
    for (int j = 1; j < 8; ++j) lmax = fmaxf(lmax, s[j]);
    lmax = wave_reduce_max(lmax);
    if (lane == 0) red_max[wave] = lmax;
    __syncthreads();
    float rmax = red_max[0];
    #pragma unroll
    for (int w = 1; w < WAVES; ++w) rmax = fmaxf(rmax, red_max[w]);

    // Guard fully-masked rows: exp(-inf - (-inf)) must not produce NaN.
    const float msafe = (rmax == NEG_INF) ? 0.0f : rmax;

    // ---- exp(s - m) with zeros at masked-out slots; local partial sum ----
    float p[8];
    float lsum = 0.0f;
    #pragma unroll
    for (int j = 0; j < 8; ++j) {
        float e = (s[j] == NEG_INF) ? 0.0f : __expf(s[j] - msafe);
        p[j] = e;
        lsum += e;
    }

    // ---- Row sum ----
    lsum = wave_reduce_sum(lsum);
    if (lane == 0) red_sum[wave] = lsum;
    __syncthreads();
    float denom = 0.0f;
    #pragma unroll
    for (int w = 0; w < WAVES; ++w) denom += red_sum[w];

    const float scale = (denom > 0.0f) ? (1.0f / denom) : 0.0f;

    // ---- Single streaming write ----
    v4f o0, o1;
    #pragma unroll
    for (int j = 0; j < 4; ++j) {
        o0[j] = p[j]     * scale;
        o1[j] = p[4 + j] * scale;
    }
    __builtin_nontemporal_store(o0, (v4f*)(o + i0));
    __builtin_nontemporal_store(o1, (v4f*)(o + i1));
}

extern "C" void kernel_launch(void* const* d_in, const int* in_sizes, int n_in,
                              void* d_out, int out_size, void* d_ws, size_t ws_size,
                              hipStream_t stream) {
    (void)n_in; (void)out_size; (void)d_ws; (void)ws_size;
    const float* feat = (const float*)d_in[0];
    const int*   od   = (const int*)d_in[1];
    float*       out  = (float*)d_out;

    const int rows = in_sizes[0] / ROW_LEN;   // 32 * 2048 = 65536 rows
    SparseSoftmax_65944927863275_kernel<<<rows, TPB, 0, stream>>>(feat, od, out);
}